// JointBilateral_58634893524997
// MI455X (gfx1250) — compile-verified
//
#include <hip/hip_runtime.h>
#include <cstdint>

// Joint bilateral (PAC) 4x upsampling, 512x512 output, batch 4.
// Memory-bound (~17 MB @ 23.3 TB/s ~ 0.73 us floor). No GEMM structure
// (<=4-tap masked weighted mean per pixel), so WMMA is inapplicable; the
// CDNA5 paths that pay are TDM tile DMA (TENSORcnt) for interior tiles and
// per-lane async global->LDS copies (ASYNCcnt) for boundary tiles.

#define H_OUT 512
#define W_OUT 512
#define H_LO  128
#define W_LO  128
#define BATCH 4

#define TW 32              // tile width  (= wave32 width, coalesced rows)
#define TH 8               // tile height (256 threads = 8 waves)
#define GW (TW + 4)        // staged guidance width  (halo +-2) = 36
#define GH (TH + 4)        // staged guidance height (halo +-2) = 12
#define U_MAX (4 * (H_LO - 1))   // 508: last valid stride-inserted position

typedef uint32_t u32x4 __attribute__((ext_vector_type(4)));
typedef uint32_t u32x8 __attribute__((ext_vector_type(8)));

// Generic pointers to LDS carry the LDS byte offset in addr[31:0]
// (CDNA5 ISA: flat->LDS aperture mapping).
__device__ __forceinline__ uint32_t lds_off_u32(const void* p) {
    return (uint32_t)(uintptr_t)p;
}

__global__ __launch_bounds__(TW * TH)
void jbu_pac_kernel(const float* __restrict__ x,
                    const float* __restrict__ guid,
                    float* __restrict__ out) {
    __shared__ float g[3][GH][GW];

    const int b  = blockIdx.z;
    const int w0 = blockIdx.x * TW;
    const int h0 = blockIdx.y * TH;
    const int tx = threadIdx.x;
    const int ty = threadIdx.y;
    const int tid = ty * TW + tx;

    const uint64_t gbase = (uint64_t)(uintptr_t)guid;

    const bool interior = (h0 - 2 >= 0) && (h0 - 2 + GH <= H_OUT) &&
                          (w0 - 2 >= 0) && (w0 - 2 + GW <= W_OUT);

    if (interior) {
        // ---- TDM fast path: one wave issues 3 tensor-tile DMAs
        //      (36x12 dwords per guidance channel) straight into LDS. ----
        if (tid < 32) {                     // wave 0 only
            u32x8 d1;
            d1[0] = 0x00020000u;            // wg_mask=0, data_size=2 (4B)
            d1[1] = (uint32_t)W_OUT << 16;  // tensor_dim0[15:0]=512 @bits79:48
            d1[2] = (uint32_t)H_OUT << 16;  // tensor_dim1[15:0]=512 @bits111:80
            d1[3] = (uint32_t)GW << 16;     // tile_dim0=36 @bits127:112
            d1[4] = (uint32_t)GH;           // tile_dim1=12, tile_dim2=0 (2D)
            d1[5] = (uint32_t)W_OUT;        // tensor_dim0_stride=512 (48b lo)
            d1[6] = 0u;                     // dim0_stride hi / dim1_stride lo16
            d1[7] = (uint32_t)(((uint64_t)H_OUT * W_OUT) >> 16); // dim1_stride hi
            #pragma unroll
            for (int c = 0; c < 3; ++c) {
                const uint64_t ga = gbase +
                    (uint64_t)((((b * 3 + c) * H_OUT + (h0 - 2)) * W_OUT +
                                (w0 - 2)) * (int)sizeof(float));
                u32x4 d0;
                d0[0] = 1u;                             // count=1, user mode
                d0[1] = lds_off_u32(&g[c][0][0]);       // LDS byte address
                d0[2] = (uint32_t)ga;                   // global_addr[31:0]
                d0[3] = (uint32_t)(ga >> 32) | (2u << 30); // addr[56:32]|type=2
                asm volatile("tensor_load_to_lds %0, %1"
                             :: "s"(d0), "s"(d1) : "memory");
            }
#if __has_builtin(__builtin_amdgcn_s_wait_tensorcnt)
            __builtin_amdgcn_s_wait_tensorcnt(0);
#else
            asm volatile("s_wait_tensorcnt 0x0" ::: "memory");
#endif
        }
        __syncthreads();
    } else {
        // ---- Boundary path: per-lane async global->LDS copies with the
        //      halo clamped to the image (clamped texels are only read by
        //      taps the validity mask rejects). ----
        for (int e = tid; e < 3 * GH * GW; e += TW * TH) {
            int c   = e / (GH * GW);
            int rem = e - c * (GH * GW);
            int r   = rem / GW;
            int col = rem - r * GW;
            int gh = h0 - 2 + r;
            int gw = w0 - 2 + col;
            gh = gh < 0 ? 0 : (gh > H_OUT - 1 ? H_OUT - 1 : gh);
            gw = gw < 0 ? 0 : (gw > W_OUT - 1 ? W_OUT - 1 : gw);
            uint32_t voff = (uint32_t)((((b * 3 + c) * H_OUT + gh) * W_OUT + gw)
                                       * sizeof(float));
            uint32_t ldsa = lds_off_u32(&g[c][r][col]);
            asm volatile("global_load_async_to_lds_b32 %0, %1, %2"
                         :: "v"(ldsa), "v"(voff), "s"(gbase)
                         : "memory");
        }
        asm volatile("s_wait_asynccnt 0x0" ::: "memory");
        __syncthreads();
    }

    // ---- Per-pixel bilateral gather: <=4 valid taps out of 5x5. ----
    const int h = h0 + ty;
    const int w = w0 + tx;
    const float c0 = g[0][ty + 2][tx + 2];
    const float c1 = g[1][ty + 2][tx + 2];
    const float c2 = g[2][ty + 2][tx + 2];

    const float* __restrict__ xb = x + b * (H_LO * W_LO);

    float acc  = 0.0f;
    float norm = 0.0f;
    #pragma unroll
    for (int i = 0; i < 5; ++i) {
        const int uh = h + i - 3;                 // stride-inserted row coord
        if (uh < 0 || uh > U_MAX || (uh & 3)) continue;
        const int m  = uh >> 2;                   // low-res row
        const int dh = i - 2;                     // guidance row offset (=uh+1-h)
        #pragma unroll
        for (int j = 0; j < 5; ++j) {
            const int uw = w + j - 3;
            if (uw < 0 || uw > U_MAX || (uw & 3)) continue;
            const int n  = uw >> 2;
            const int dw = j - 2;
            // Color term: reference scales guidance by 0.5 -> 0.25*raw diff^2.
            const float d0 = g[0][ty + 2 + dh][tx + 2 + dw] - c0;
            const float d1 = g[1][ty + 2 + dh][tx + 2 + dw] - c1;
            const float d2 = g[2][ty + 2 + dh][tx + 2 + dw] - c2;
            // Spatial term: (0.1*dh)^2 + (0.1*dw)^2 (guidance is in-bounds
            // for every valid tap: row = uh+1 in [1,509]).
            const float dist = 0.25f * (d0 * d0 + d1 * d1 + d2 * d2)
                             + 0.01f * (float)(dh * dh + dw * dw);
            const float wt = __expf(-0.5f * dist);
            acc  += wt * xb[m * W_LO + n];
            norm += wt;
        }
    }

    const float denom = norm + (norm == 0.0f ? 1.0f : 0.0f);
    out[(b * H_OUT + h) * W_OUT + w] = acc / denom;
}

extern "C" void kernel_launch(void* const* d_in, const int* in_sizes, int n_in,
                              void* d_out, int out_size, void* d_ws, size_t ws_size,
                              hipStream_t stream) {
    const float* x    = (const float*)d_in[0];   // (4,1,128,128) f32
    const float* guid = (const float*)d_in[1];   // (4,3,512,512) f32
    // d_in[2] is the all-ones conv-transpose weight; it folds to 1.0 exactly.
    float* out = (float*)d_out;                  // (4,1,512,512) f32

    dim3 block(TW, TH);
    dim3 grid(W_OUT / TW, H_OUT / TH, BATCH);
    jbu_pac_kernel<<<grid, block, 0, stream>>>(x, guid, out);
}